// MeanAggregator_61392262529195
// MI455X (gfx1250) — compile-verified
//
#include <hip/hip_runtime.h>
#include <stdint.h>

// MeanAggregator: out[b, :] = mean_k features[neigh_idx[b,k], :]
//   features: [V=200000, D=256] f32   (204.8 MB, ~fits the 192MB L2)
//   neigh_idx: [B=100000, K=10] int32
//   out: [B, 256] f32                 (102.4 MB, write-once -> NT stores)
//
// Memory-bound gather (AI ~0.23 flop/byte; HBM floor ~50us at 23.3 TB/s).
// One 64-thread block per query row. Per neighbor row the 64 lanes form one
// contiguous 1KB burst issued via CDNA5 async global->LDS b128 gathers
// (ASYNCcnt path, zero VGPR pressure for staged data). Indices are
// block-uniform -> scalar s_load on the KMcnt path. Async completions are
// in-order per wave, so a split wait overlaps the LDS reduction of the first
// 5 rows with the in-flight tail.

#define K_NEIGH 10
#define DIM 256
#define THREADS 64   // DIM/4 lanes; each lane owns 4 consecutive floats (16B)

typedef float v4f __attribute__((ext_vector_type(4)));

__global__ __launch_bounds__(THREADS)
void mean_agg_async_kernel(const float* __restrict__ features,
                           const int*   __restrict__ neigh_idx,
                           float* __restrict__ out,
                           int B)
{
    __shared__ float tile[K_NEIGH][DIM];   // 10 KB staging for the 10 gathered rows

    const int b = blockIdx.x;
    if (b >= B) return;

    const int      tid       = threadIdx.x;         // 0..63
    const unsigned lane_byte = (unsigned)tid * 16u; // 16 bytes (float4) per lane

    // Block-uniform index row -> scalar (s_load) fetches.
    const int* __restrict__ row = neigh_idx + (size_t)b * K_NEIGH;

    int idx[K_NEIGH];
#pragma unroll
    for (int k = 0; k < K_NEIGH; ++k) idx[k] = row[k];

    // Issue all K async gathers before waiting: 10 in-flight 16B/lane requests;
    // each neighbor row is a fully coalesced 1KB burst across the block.
#pragma unroll
    for (int k = 0; k < K_NEIGH; ++k) {
        // byte offset of this lane's 16B chunk within the feature table
        unsigned voff = (unsigned)idx[k] * (DIM * 4u) + lane_byte;
        // LDS byte offset: low 32 bits of generic pointer == LDS address
        unsigned lds_off = (unsigned)(uintptr_t)(&tile[k][0]) + lane_byte;
        asm volatile("global_load_async_to_lds_b128 %0, %1, %2"
                     :
                     : "v"(lds_off), "v"(voff), "s"(features)
                     : "memory");
    }

    float4 acc = make_float4(0.f, 0.f, 0.f, 0.f);

    // Async loads complete in order per wave: asynccnt<=5 => rows 0..4 landed.
    asm volatile("s_wait_asynccnt 5" ::: "memory");
#pragma unroll
    for (int k = 0; k < 5; ++k) {
        const float4 v = *(const float4*)(&tile[k][tid * 4]);
        acc.x += v.x; acc.y += v.y; acc.z += v.z; acc.w += v.w;
    }

    asm volatile("s_wait_asynccnt 0" ::: "memory");
#pragma unroll
    for (int k = 5; k < K_NEIGH; ++k) {
        const float4 v = *(const float4*)(&tile[k][tid * 4]);
        acc.x += v.x; acc.y += v.y; acc.z += v.z; acc.w += v.w;
    }

    const float s = 1.0f / (float)K_NEIGH;
    v4f r;
    r.x = acc.x * s; r.y = acc.y * s; r.z = acc.z * s; r.w = acc.w * s;

    // Write-once output: non-temporal store keeps the gather table resident in L2.
    __builtin_nontemporal_store(r, (v4f*)(out + (size_t)b * DIM + tid * 4));
}

extern "C" void kernel_launch(void* const* d_in, const int* in_sizes, int n_in,
                              void* d_out, int out_size, void* d_ws, size_t ws_size,
                              hipStream_t stream)
{
    const float* features  = (const float*)d_in[0];   // [V, 256] f32
    const int*   neigh_idx = (const int*)d_in[1];     // [B, 10] int32
    float*       out       = (float*)d_out;           // [B, 256] f32

    const int B = in_sizes[1] / K_NEIGH;

    dim3 grid((unsigned)B), block(THREADS);
    mean_agg_async_kernel<<<grid, block, 0, stream>>>(features, neigh_idx, out, B);
}